// ScaledDotProductAttention_68607807586645
// MI455X (gfx1250) — compile-verified
//
#include <hip/hip_runtime.h>
#include <cstddef>

typedef __attribute__((ext_vector_type(2))) float v2f;
typedef __attribute__((ext_vector_type(8))) float v8f;

#define S_LEN   2048
#define D_DIM   64
#define H_NUM   16
#define B_NUM   2
#define QT      16              // q rows per block
#define WAVES   4
#define THREADS (WAVES * 32)
#define KTILES  (S_LEN / 16)    // 128 key tiles of 16
#define SCALE   0.125f          // 1/sqrt(64)

// D = A(16x4) * B(4x16) + C, fp32, wave32 WMMA
static __device__ __forceinline__ v8f wmma4(v2f a, v2f b, v8f c) {
  return __builtin_amdgcn_wmma_f32_16x16x4_f32(false, a, false, b,
                                               (short)0, c, false, false);
}

__global__ __launch_bounds__(THREADS)
void attn_fused_kernel(const float* __restrict__ Q,
                       const float* __restrict__ Kg,
                       const float* __restrict__ V,
                       const int*   __restrict__ Mask,
                       float* __restrict__ Out,
                       float* __restrict__ Pattn)
{
  __shared__ __align__(16) float sc[QT * S_LEN];          // 128 KB scores->e
  __shared__ __align__(16) float redmax[WAVES * QT * 16]; // 4 KB per-wave max
  __shared__ float red[QT * 8];                           // row sum partials
  __shared__ float rowinv[QT];                            // 1/rowsum

  const int tid  = threadIdx.x;
  const int lane = tid & 31;
  const int wave = tid >> 5;

  const int blk   = blockIdx.x;
  const int qtile = blk & (KTILES - 1);   // 128 q-tiles per head
  const int bh    = blk >> 7;             // 0..31 (b*H + h)
  const int b     = bh >> 4;
  const int q0    = qtile * QT;

  const float* Qh = Q    + (size_t)bh * S_LEN * D_DIM;
  const float* Kh = Kg   + (size_t)bh * S_LEN * D_DIM;
  const float* Vh = V    + (size_t)bh * S_LEN * D_DIM;
  const int*   Mh = Mask + (size_t)b  * S_LEN * S_LEN;  // (B,1,S,S)

  // fragment geometry (ISA 7.12.2, 32-bit 16x4 A / 4x16 B / 16x16 C)
  const int mrow = lane & 15;   // A: M row / B,C: N col
  const int half = lane >> 4;   // 0: K={0,1}, C rows M=i ; 1: K={2,3}, M=i+8
  const int kb   = half * 2;

  // ---- preload Q A-fragments, scale folded in (attn = (QK^T)*scale) ----
  v2f qa[16];
  {
    const float* qrow = Qh + (size_t)(q0 + mrow) * D_DIM;
#pragma unroll
    for (int ks = 0; ks < 16; ++ks)
      qa[ks] = (*(const v2f*)(qrow + 4 * ks + kb)) * SCALE;  // 8B aligned
  }

  // ---- phase 1: raw scores S = (scale*Q) K^T into LDS + running row max ----
  // uniform trip count -> scalar loop, EXEC stays all-ones around WMMA
  v8f mx8;
#pragma unroll
  for (int i = 0; i < 8; ++i) mx8[i] = -3.4e38f;

#pragma unroll 1
  for (int t = 0; t < KTILES / WAVES; ++t) {
    const int kt = t * WAVES + wave;
    const int k0 = kt * 16;
    const float* krow = Kh + (size_t)(k0 + mrow) * D_DIM;

    // prefetch the K row this lane needs next iteration
    const int ktn = (kt + WAVES < KTILES) ? (kt + WAVES) : kt;
    __builtin_prefetch(Kh + (size_t)(ktn * 16 + mrow) * D_DIM + kb, 0, 1);

    v8f c = {};
#pragma unroll
    for (int ks = 0; ks < 16; ++ks) {
      v2f bk = *(const v2f*)(krow + 4 * ks + kb);   // B[k][n] = K[n][d]
      c = wmma4(qa[ks], bk, c);
    }
#pragma unroll
    for (int i = 0; i < 8; ++i) {
      const int rm = i + half * 8;                  // C: M index
      sc[rm * S_LEN + k0 + mrow] = c[i];
      mx8[i] = fmaxf(mx8[i], c[i]);                 // raw-score shift constant
    }
  }
  // one writer per slot: wave-private max slab, no atomics/init needed
#pragma unroll
  for (int i = 0; i < 8; ++i) {
    const int rm = i + half * 8;
    redmax[wave * (QT * 16) + rm * 16 + mrow] = mx8[i];
  }
  __syncthreads();

  // ---- phase 2: rowmax from 4KB slab, then ONE fused mask+exp+sum pass ----
  const int r   = tid >> 3;
  const int sub = tid & 7;

  float rowmax = -3.4e38f;
  {
    const float4* rm4 = reinterpret_cast<const float4*>(redmax);
#pragma unroll
    for (int w = 0; w < WAVES; ++w)
#pragma unroll
      for (int j = 0; j < 4; ++j) {
        const float4 f = rm4[w * (QT * 4) + r * 4 + j];
        rowmax = fmaxf(rowmax, fmaxf(fmaxf(f.x, f.y), fmaxf(f.z, f.w)));
      }
  }
  // NOTE: rowmax is max of RAW scores (>= masked max) -> softmax identical;
  // masked entries become exact 0 (reference's exp(-1e9-max) underflows to 0).
  float4* row4 = reinterpret_cast<float4*>(sc + r * S_LEN);
  const int4* mrow4 = reinterpret_cast<const int4*>(Mh + (size_t)(q0 + r) * S_LEN);

  float sum = 0.f;
  for (int i = sub; i < S_LEN / 4; i += 8) {
    float4 f = row4[i];
    const int4 m = mrow4[i];                  // coalesced 128B mask reads
    f.x = (m.x > 0) ? 0.f : __expf(f.x - rowmax);
    f.y = (m.y > 0) ? 0.f : __expf(f.y - rowmax);
    f.z = (m.z > 0) ? 0.f : __expf(f.z - rowmax);
    f.w = (m.w > 0) ? 0.f : __expf(f.w - rowmax);
    row4[i] = f;                              // keep unnormalized e in LDS
    sum += (f.x + f.y) + (f.z + f.w);
  }
  red[r * 8 + sub] = sum;
  __syncthreads();                            // e[] + sum partials visible
  float rowsum = 0.f;
#pragma unroll
  for (int i = 0; i < 8; ++i) rowsum += red[r * 8 + i];
  if (sub == 0) rowinv[r] = 1.0f / rowsum;
  __syncthreads();                            // rowinv visible to all

  // ---- phase 3: p_attn = e * rowinv, streamed out as b128 stores ----
  // (no LDS write-back, no barrier: sc stays unnormalized-e for phase 4,
  //  stores drain in background while phase 4 WMMAs run)
  {
    const float4* sc4 = reinterpret_cast<const float4*>(sc);
    float4* Pout4 = reinterpret_cast<float4*>(
        Pattn + ((size_t)bh * S_LEN + q0) * S_LEN);
    for (int i4 = tid; i4 < QT * S_LEN / 4; i4 += THREADS) {
      const int   rr  = i4 >> 9;              // (i4*4)/2048
      const float inv = rowinv[rr];
      float4 e = sc4[i4];
      e.x *= inv; e.y *= inv; e.z *= inv; e.w *= inv;
      Pout4[i4] = e;                          // contiguous == (rr, col) layout
    }
  }

  // ---- phase 4: out = (e V) * rowinv ; each wave owns one 16-col slab ----
  const int col0 = wave * 16;
  v8f o = {};
#pragma unroll 2
  for (int kt = 0; kt < KTILES; ++kt) {
#pragma unroll
    for (int ks = 0; ks < 4; ++ks) {
      const int kk0 = kt * 16 + ks * 4;
      v2f a = *(const v2f*)(&sc[mrow * S_LEN + kk0 + kb]);  // A[m][k]=e
      v2f bv;
      bv.x = Vh[(size_t)(kk0 + kb)     * D_DIM + col0 + mrow];
      bv.y = Vh[(size_t)(kk0 + kb + 1) * D_DIM + col0 + mrow];
      o = wmma4(a, bv, o);
    }
  }
  float* Oh = Out + ((size_t)bh * S_LEN + q0) * D_DIM + col0;
#pragma unroll
  for (int i = 0; i < 8; ++i) {
    const int rm = i + half * 8;
    Oh[(size_t)rm * D_DIM + mrow] = o[i] * rowinv[rm];  // fold normalization
  }
}

extern "C" void kernel_launch(void* const* d_in, const int* in_sizes, int n_in,
                              void* d_out, int out_size, void* d_ws, size_t ws_size,
                              hipStream_t stream) {
  (void)in_sizes; (void)n_in; (void)out_size; (void)d_ws; (void)ws_size;
  const float* Q = (const float*)d_in[0];
  const float* K = (const float*)d_in[1];
  const float* V = (const float*)d_in[2];
  const int*   M = (const int*)d_in[3];

  float* out   = (float*)d_out;                                   // (B,H,S,D)
  float* pattn = out + (size_t)B_NUM * H_NUM * S_LEN * D_DIM;     // (B,H,S,S)

  dim3 grid(B_NUM * H_NUM * (S_LEN / QT));   // 4096 blocks
  attn_fused_kernel<<<grid, THREADS, 0, stream>>>(Q, K, V, M, out, pattn);
}